// KeypointLoss_5738076307628
// MI455X (gfx1250) — compile-verified
//
#include <hip/hip_runtime.h>
#include <hip/hip_bf16.h>
#include <math.h>

#define S_ 4
#define B_ 16
#define K_ 6
#define NCLASS_ 7
#define C_ (2 * K_ + NCLASS_)   // 19
#define H_ 128
#define W_ 128
#define HW_ (H_ * W_)           // 16384
#define SMOOTH_ 0.1f
#define SBK_ (S_ * B_ * K_)     // 384

typedef __attribute__((ext_vector_type(2))) float v2f;
typedef __attribute__((ext_vector_type(8))) float v8f;

__device__ __forceinline__ float bce_logits(float x, float t) {
    return fmaxf(x, 0.0f) - x * t + log1pf(expf(-fabsf(x)));
}

__device__ __forceinline__ float waveReduceSum(float v) {
    #pragma unroll
    for (int off = 16; off > 0; off >>= 1)
        v += __shfl_down(v, off, 32);
    return v;
}

__device__ __forceinline__ unsigned long long waveReduceMaxU64(unsigned long long v) {
    #pragma unroll
    for (int off = 16; off > 0; off >>= 1) {
        unsigned long long o = __shfl_down(v, off, 32);
        v = (o > v) ? o : v;
    }
    return v;
}

// Sum 32 per-lane f32 values across the wave using the CDNA5 matrix pipe.
// Lane L contributes lane_val in A[M=L%16][K = (L<16)?0:2] of a 16x4 f32
// A-matrix (VGPR0 = lane_val, VGPR1 = 0). B = all-ones 4x16, so
// D[m][n] = sum_k A[m][k] (row sums). Summing the 8 D VGPRs in a lane gives
// a partial column sum; column 0 = lanes 0 and 16 combined give the total.
// Must be executed by a full wave (EXEC all ones).
__device__ __forceinline__ float wmma_sum32(float lane_val) {
    v2f a;  a[0] = lane_val; a[1] = 0.0f;
    v2f bm; bm[0] = 1.0f;    bm[1] = 1.0f;
    v8f c = {};
    v8f d = __builtin_amdgcn_wmma_f32_16x16x4_f32(
        /*neg_a=*/false, a, /*neg_b=*/false, bm,
        /*c_mod=*/(short)0, c, /*reuse_a=*/false, /*reuse_b=*/false);
    float t = d[0] + d[1] + d[2] + d[3] + d[4] + d[5] + d[6] + d[7];
    return __shfl(t, 0, 32) + __shfl(t, 16, 32);
}

// Kernel 1: one block per (s,b,k). Streams 4 channels (pred-hp, pred-mask,
// heatmap, mask) once with float4 loads; produces per-(s,b,k) squared-error
// sum, mask-BCE sum, and argmax index of pred*mask.
__global__ __launch_bounds__(1024)
void keyloss_stream_kernel(const float* __restrict__ preds,
                           const float* __restrict__ heat,
                           const float* __restrict__ masks,
                           float* __restrict__ ws_sq,
                           float* __restrict__ ws_bce,
                           unsigned int* __restrict__ ws_idx) {
    const int g  = blockIdx.x;      // s*B*K + b*K + k
    const int k  = g % K_;
    const int sb = g / K_;          // s*B + b

    const float4* hp4 = (const float4*)(preds + ((size_t)sb * C_ + k) * HW_);
    const float4* mp4 = (const float4*)(preds + ((size_t)sb * C_ + K_ + k) * HW_);
    const float4* hm4 = (const float4*)(heat  + ((size_t)sb * K_ + k) * HW_);
    const float4* mk4 = (const float4*)(masks + ((size_t)sb * K_ + k) * HW_);

    __shared__ float s_sq[32];
    __shared__ float s_bce[32];
    __shared__ unsigned long long s_amax;

    const int tid = threadIdx.x;
    if (tid == 0) s_amax = 0ULL;
    __syncthreads();

    float sq = 0.0f, bce = 0.0f;
    float bestV = -INFINITY;
    int   bestI = 0;

    const float tgt_bias = SMOOTH_ / (float)W_;

    for (int i = tid; i < HW_ / 4; i += 1024) {
        float4 p  = hp4[i];
        float4 m  = mk4[i];
        float4 hh = hm4[i];
        float4 xm = mp4[i];
        float pa[4] = {p.x,  p.y,  p.z,  p.w};
        float ma[4] = {m.x,  m.y,  m.z,  m.w};
        float ha[4] = {hh.x, hh.y, hh.z, hh.w};
        float xa[4] = {xm.x, xm.y, xm.z, xm.w};
        #pragma unroll
        for (int c = 0; c < 4; ++c) {
            float hv = pa[c] * ma[c];
            float dd = hv - ha[c];
            sq = fmaf(dd, dd, sq);
            int idx = i * 4 + c;                 // ascending per thread
            if (hv > bestV) { bestV = hv; bestI = idx; }
            float x  = xa[c];
            float tt = fmaf(ma[c], 1.0f - SMOOTH_, tgt_bias);
            bce += bce_logits(x, tt);
        }
    }

    // pack argmax as monotone key; ~idx so equal values pick the smaller index
    unsigned int vb = __float_as_uint(bestV);
    vb = (vb & 0x80000000u) ? ~vb : (vb | 0x80000000u);
    unsigned long long key =
        ((unsigned long long)vb << 32) | (unsigned int)(~(unsigned int)bestI);

    sq  = waveReduceSum(sq);
    bce = waveReduceSum(bce);
    key = waveReduceMaxU64(key);

    const int lane = tid & 31;
    const int wave = tid >> 5;
    if (lane == 0) {
        s_sq[wave]  = sq;
        s_bce[wave] = bce;
        atomicMax(&s_amax, key);                 // ds_max_u64 in LDS
    }
    __syncthreads();

    if (tid < 32) {                              // wave 0, EXEC all ones
        float tsq  = wmma_sum32(s_sq[tid]);
        float tbce = wmma_sum32(s_bce[tid]);
        if (tid == 0) {
            ws_sq[g]  = tsq;
            ws_bce[g] = tbce;
            ws_idx[g] = ~(unsigned int)(s_amax & 0xFFFFFFFFULL);
        }
    }
}

// Kernel 2: one block per (s,b). Gathers the 6x7 class logits at the argmax
// indices (lines still hot in L2), computes labels BCE, folds per-k partials,
// writes the three transposed (B,S) outputs.
__global__ __launch_bounds__(64)
void keyloss_final_kernel(const float* __restrict__ preds,
                          const float* __restrict__ labels,
                          const float* __restrict__ ws_sq,
                          const float* __restrict__ ws_bce,
                          const unsigned int* __restrict__ ws_idx,
                          float* __restrict__ out) {
    const int sb = blockIdx.x;        // s*B + b
    const int s  = sb / B_;
    const int b  = sb % B_;
    const int tid = threadIdx.x;

    __shared__ float s_lab[2];

    float l = 0.0f;
    if (tid < K_ * NCLASS_) {
        int k = tid / NCLASS_;
        int c = tid % NCLASS_;
        unsigned int idx = ws_idx[sb * K_ + k];
        float x = preds[((size_t)sb * C_ + 2 * K_ + c) * HW_ + idx];
        float t = fmaf(labels[((size_t)b * K_ + k) * NCLASS_ + c],
                       1.0f - SMOOTH_, SMOOTH_ / (float)NCLASS_);
        l = bce_logits(x, t);
    }
    l = waveReduceSum(l);
    if ((tid & 31) == 0) s_lab[tid >> 5] = l;
    __syncthreads();

    if (tid == 0) {
        float lab = (s_lab[0] + s_lab[1]) / (float)K_;
        float comb = 0.0f, mbce = 0.0f;
        #pragma unroll
        for (int k = 0; k < K_; ++k) {
            comb += ws_sq[sb * K_ + k];
            mbce += ws_bce[sb * K_ + k];
        }
        comb /= (float)K_;
        mbce /= (float)K_;
        // outputs are (B,S) row-major, concatenated: combined, labels, masks
        out[b * S_ + s]                 = comb;
        out[S_ * B_ + b * S_ + s]       = lab;
        out[2 * S_ * B_ + b * S_ + s]   = mbce;
    }
}

extern "C" void kernel_launch(void* const* d_in, const int* in_sizes, int n_in,
                              void* d_out, int out_size, void* d_ws, size_t ws_size,
                              hipStream_t stream) {
    const float* preds  = (const float*)d_in[0];   // (S,B,C,H,W)
    const float* heat   = (const float*)d_in[1];   // (S,B,K,H,W)
    const float* labels = (const float*)d_in[2];   // (B,K,NCLASS)
    const float* masks  = (const float*)d_in[3];   // (S,B,K,H,W)
    float* out = (float*)d_out;

    float*        ws_sq  = (float*)d_ws;
    float*        ws_bce = ws_sq + SBK_;
    unsigned int* ws_idx = (unsigned int*)(ws_bce + SBK_);

    keyloss_stream_kernel<<<SBK_, 1024, 0, stream>>>(preds, heat, masks,
                                                     ws_sq, ws_bce, ws_idx);
    keyloss_final_kernel<<<S_ * B_, 64, 0, stream>>>(preds, labels,
                                                     ws_sq, ws_bce, ws_idx, out);
}